// TriMipEncoding_26379689132063
// MI455X (gfx1250) — compile-verified
//
#include <hip/hip_runtime.h>
#include <stdint.h>

#define PLANE 512
#define FDIM  16

typedef __attribute__((ext_vector_type(4))) float f4;

__device__ __forceinline__ int iclamp(int v, int lo, int hi) {
    return v < lo ? lo : (v > hi ? hi : v);
}

// 256 threads/block = 8 wave32.
// Lane layout: 4 lanes per point (one float4 of the 16 channels each),
// 8 points per wave, 64 points per block.
__global__ __launch_bounds__(256)
void TriMipEncoding_kernel(const float* __restrict__ x,
                           const float* __restrict__ fm,
                           float* __restrict__ out,
                           int n) {
    __shared__ float sx[64 * 3];  // this block's 64 points * (x,y,z)

    const int tid       = threadIdx.x;
    const int block_pt0 = blockIdx.x * 64;

    // ---- Stage the block's coordinates into LDS with CDNA5 async copy ----
    // 192 floats -> exactly waves 0..5 fully active (wave32).
    if (tid < 192 && (block_pt0 * 3 + tid) < n * 3) {
        uint32_t lds_off = (uint32_t)(uintptr_t)(&sx[tid]);        // low 32b of generic = LDS offset
        uint32_t goff    = (uint32_t)((block_pt0 * 3 + tid) * 4);  // byte offset into x
        asm volatile("global_load_async_to_lds_b32 %0, %1, %2"
                     :: "v"(lds_off), "v"(goff), "s"(x)
                     : "memory");
    }
    asm volatile("s_wait_asynccnt 0" ::: "memory");
    __syncthreads();

    const int p_local = tid >> 2;   // point within block: 0..63
    const int cg      = tid & 3;    // which float4 of the 16 channels
    const int p       = block_pt0 + p_local;
    if (p >= n) return;

    const float xc = sx[p_local * 3 + 0];
    const float yc = sx[p_local * 3 + 1];
    const float zc = sx[p_local * 3 + 2];

    // plane 0: (u,v) = (y,z); plane 1: (x,z); plane 2: (x,y)
    const float us[3] = { yc, xc, xc };
    const float vs[3] = { zc, zc, yc };

    const f4* __restrict__ fmv = (const f4*)fm;

#pragma unroll
    for (int pl = 0; pl < 3; ++pl) {
        const float u = us[pl] * (float)PLANE - 0.5f;
        const float v = vs[pl] * (float)PLANE - 0.5f;
        const float i0f = floorf(u);
        const float j0f = floorf(v);
        const float fu = u - i0f;
        const float fv = v - j0f;
        const int i0 = iclamp((int)i0f,     0, PLANE - 1);
        const int i1 = iclamp((int)i0f + 1, 0, PLANE - 1);
        const int j0 = iclamp((int)j0f,     0, PLANE - 1);
        const int j1 = iclamp((int)j0f + 1, 0, PLANE - 1);

        // f4-unit offsets: ((pl*512 + j)*512 + i) * (16/4) + cg
        const int row0 = (pl * PLANE + j0) * PLANE;
        const int row1 = (pl * PLANE + j1) * PLANE;

        // 4 x global_load_b128; fm stays resident in the 192MB L2 (RT hint).
        const f4 t00 = fmv[(row0 + i0) * 4 + cg];
        const f4 t01 = fmv[(row0 + i1) * 4 + cg];
        const f4 t10 = fmv[(row1 + i0) * 4 + cg];
        const f4 t11 = fmv[(row1 + i1) * 4 + cg];

        const float w00 = (1.0f - fv) * (1.0f - fu);
        const float w01 = (1.0f - fv) * fu;
        const float w10 = fv * (1.0f - fu);
        const float w11 = fv * fu;

        f4 r = t00 * w00 + t01 * w01 + t10 * w10 + t11 * w11;

        // Non-temporal b128 store: out is a 154MB one-shot stream; keep it
        // from evicting the 48MB tri-plane texture out of L2.
        __builtin_nontemporal_store(r, (f4*)(out + (size_t)p * 48 + pl * 16 + cg * 4));
    }
}

extern "C" void kernel_launch(void* const* d_in, const int* in_sizes, int n_in,
                              void* d_out, int out_size, void* d_ws, size_t ws_size,
                              hipStream_t stream) {
    const float* x  = (const float*)d_in[0];   // [N,3] f32
    const float* fm = (const float*)d_in[1];   // [3,512,512,16] f32
    float* out      = (float*)d_out;           // [N,48] f32

    const int n = in_sizes[0] / 3;
    const int blocks = (n + 63) / 64;          // 64 points per 256-thread block
    TriMipEncoding_kernel<<<blocks, 256, 0, stream>>>(x, fm, out, n);
}